// Nonbonded_46969762349276
// MI455X (gfx1250) — compile-verified
//
#include <hip/hip_runtime.h>

#define THREADS 256
#define BLOCKS  2048
#define WAVES_PER_BLOCK (THREADS / 32)

typedef float v2f __attribute__((ext_vector_type(2)));
typedef float v8f __attribute__((ext_vector_type(8)));

// 32-byte packed atom record: one L2 cacheline-friendly gather per atom.
struct AtomPacked { float x, y, z, q, sig, eps, pad0, pad1; };

// ---------------------------------------------------------------------------
// Prep: pack per-atom data (coords float3 + q + sigma + eps) into 32B structs
// in workspace so the hot loop does 2x b128 gathers per atom instead of 6
// scattered b32 gathers across 4 arrays.
// ---------------------------------------------------------------------------
__global__ void nb_prep_kernel(const float* __restrict__ coords,
                               const float* __restrict__ sigma,
                               const float* __restrict__ epsilon,
                               const float* __restrict__ charges,
                               AtomPacked* __restrict__ atoms, int n_atoms) {
    int a = blockIdx.x * blockDim.x + threadIdx.x;
    if (a >= n_atoms) return;
    float4 lo, hi;
    lo.x = coords[3 * a + 0];
    lo.y = coords[3 * a + 1];
    lo.z = coords[3 * a + 2];
    lo.w = charges[a];
    hi.x = sigma[a];
    hi.y = epsilon[a];
    hi.z = 0.0f;
    hi.w = 0.0f;
    float4* dst = (float4*)&atoms[a];
    dst[0] = lo;   // global_store_b128
    dst[1] = hi;   // global_store_b128
}

// ---------------------------------------------------------------------------
// Wave-level sum on the matrix unit: V_WMMA_F32_16X16X4_F32 with A = ones.
// B vgpr0 = per-lane value v  -> rows K=0 (lanes 0-15) and K=2 (lanes 16-31)
// B vgpr1 = 0                 -> rows K=1, K=3
// D[m][n] = v[n] + v[n+16]  (16 column sums, duplicated across both lane
// halves of D vgpr0). Finish with 4 xor ds_swizzles over the 16-lane group.
// EXEC is all-1s here (no divergent exits before this point).
// ---------------------------------------------------------------------------
__device__ __forceinline__ float wave_reduce_sum(float v) {
#if defined(__HIP_DEVICE_COMPILE__) && __has_builtin(__builtin_amdgcn_wmma_f32_16x16x4_f32)
    v2f a; a.x = 1.0f; a.y = 1.0f;
    v2f b; b.x = v;    b.y = 0.0f;
    v8f c = {};
    c = __builtin_amdgcn_wmma_f32_16x16x4_f32(
        /*neg_a=*/false, a, /*neg_b=*/false, b,
        /*c_mod=*/(short)0, c, /*reuse_a=*/false, /*reuse_b=*/false);
    float col = c[0];                 // colsum[lane % 16] in every lane
#if __has_builtin(__builtin_amdgcn_ds_swizzle)
    // ds_swizzle group-of-32: offset[4:0]=and, [9:5]=or, [14:10]=xor
    col += __int_as_float(__builtin_amdgcn_ds_swizzle(__float_as_int(col), 0x041F)); // ^1
    col += __int_as_float(__builtin_amdgcn_ds_swizzle(__float_as_int(col), 0x081F)); // ^2
    col += __int_as_float(__builtin_amdgcn_ds_swizzle(__float_as_int(col), 0x101F)); // ^4
    col += __int_as_float(__builtin_amdgcn_ds_swizzle(__float_as_int(col), 0x201F)); // ^8
#else
    col += __shfl_xor(col, 1, 32);
    col += __shfl_xor(col, 2, 32);
    col += __shfl_xor(col, 4, 32);
    col += __shfl_xor(col, 8, 32);
#endif
    return col;                       // full wave total in every lane
#else
    // host-pass / fallback: plain shuffle tree (never executed on device
    // when the WMMA builtin is available)
    for (int m = 16; m >= 1; m >>= 1) v += __shfl_xor(v, m, 32);
    return v;
#endif
}

// ---------------------------------------------------------------------------
// Main pair kernel: grid-stride over the 128 MiB pairs stream (b128 loads +
// branch-free clamped prefetch of the next stride into near caches),
// L2-resident random gathers of packed atoms, fp32 energy math, per-block
// double partial to workspace (deterministic). Templated on PACKED so the
// fallback gather path generates zero code in the packed instantiation.
// ---------------------------------------------------------------------------
template <bool PACKED>
__global__ void nb_pair_kernel(const longlong2* __restrict__ pairs,
                               const AtomPacked* __restrict__ atoms,
                               const float* __restrict__ coords,
                               const float* __restrict__ sigma,
                               const float* __restrict__ epsilon,
                               const float* __restrict__ charges,
                               const float* __restrict__ box,
                               const float* __restrict__ coul_ptr,
                               const float* __restrict__ cutoff_ptr,
                               long long npairs,
                               double* __restrict__ partials) {
    const float Lx = box[0], Ly = box[4], Lz = box[8];
    const float iLx = 1.0f / Lx, iLy = 1.0f / Ly, iLz = 1.0f / Lz;
    const float coul    = coul_ptr[0];
    const float cutoff  = cutoff_ptr[0];
    const float inv_cut = 1.0f / cutoff;
    const float cut2    = cutoff * cutoff;

    const long long stride = (long long)gridDim.x * blockDim.x;
    const long long gtid   = (long long)blockIdx.x * blockDim.x + threadIdx.x;

    float acc = 0.0f;

#pragma unroll 4
    for (long long p = gtid; p < npairs; p += stride) {
        // Branch-free prefetch of next iteration's pair line into near caches
        // (address clamped in-bounds -> safe even as non-speculative).
        long long pf = p + stride;
        pf = (pf < npairs) ? pf : p;
        __builtin_prefetch((const void*)&pairs[pf], 0, 3);

        longlong2 pr = pairs[p];          // coalesced global_load_b128
        int i = (int)pr.x;
        int j = (int)pr.y;

        float xi, yi, zi, qi, si, ei;
        float xj, yj, zj, qj, sj, ej;
        if constexpr (PACKED) {
            const float4* ai = (const float4*)&atoms[i];
            const float4* aj = (const float4*)&atoms[j];
            float4 ai0 = ai[0], ai1 = ai[1];   // two b128 in one cacheline
            float4 aj0 = aj[0], aj1 = aj[1];
            xi = ai0.x; yi = ai0.y; zi = ai0.z; qi = ai0.w;
            si = ai1.x; ei = ai1.y;
            xj = aj0.x; yj = aj0.y; zj = aj0.z; qj = aj0.w;
            sj = aj1.x; ej = aj1.y;
        } else {
            xi = coords[3 * i + 0]; yi = coords[3 * i + 1]; zi = coords[3 * i + 2];
            xj = coords[3 * j + 0]; yj = coords[3 * j + 1]; zj = coords[3 * j + 2];
            qi = charges[i]; qj = charges[j];
            si = sigma[i];   sj = sigma[j];
            ei = epsilon[i]; ej = epsilon[j];
        }

        // minimum-image: jnp.round is round-half-even -> rintf (v_rndne_f32)
        float dx = xi - xj; dx -= rintf(dx * iLx) * Lx;
        float dy = yi - yj; dy -= rintf(dy * iLy) * Ly;
        float dz = zi - zj; dz -= rintf(dz * iLz) * Lz;

        float r2    = dx * dx + dy * dy + dz * dz;
        float inv_r = rsqrtf(r2);

        float e_c = coul * qi * qj * (inv_r - inv_cut);

        float sig  = 0.5f * (si + sj);
        float epsv = sqrtf(ei * ej);
        float sr   = sig * inv_r;
        float sr2  = sr * sr;
        float sr6  = sr2 * sr2 * sr2;
        float e_l  = 4.0f * epsv * (sr6 * sr6 - sr6);

        float e = e_c + e_l;
        acc += (r2 < cut2) ? e : 0.0f;    // r < cutoff  <=>  r2 < cutoff^2
    }

    // wave reduction on the matrix unit (EXEC all-1s: no early exits)
    float wtot = wave_reduce_sum(acc);

    __shared__ float wsum[WAVES_PER_BLOCK];
    const int lane = threadIdx.x & 31;
    const int wave = threadIdx.x >> 5;
    if (lane == 0) wsum[wave] = wtot;
    __syncthreads();

    if (threadIdx.x == 0) {
        double s = 0.0;
        for (int w = 0; w < WAVES_PER_BLOCK; ++w) s += (double)wsum[w];
        partials[blockIdx.x] = s;          // fixed order -> deterministic
    }
}

// ---------------------------------------------------------------------------
// Deterministic fixed-order tree reduction of the per-block double partials.
// ---------------------------------------------------------------------------
__global__ void nb_reduce_kernel(const double* __restrict__ partials, int n,
                                 float* __restrict__ out) {
    __shared__ double sdata[THREADS];
    double s = 0.0;
    for (int k = threadIdx.x; k < n; k += THREADS) s += partials[k];
    sdata[threadIdx.x] = s;
    __syncthreads();
    for (int off = THREADS / 2; off > 0; off >>= 1) {
        if (threadIdx.x < off) sdata[threadIdx.x] += sdata[threadIdx.x + off];
        __syncthreads();
    }
    if (threadIdx.x == 0) out[0] = (float)sdata[0];
}

// ---------------------------------------------------------------------------
// Launch
// ---------------------------------------------------------------------------
extern "C" void kernel_launch(void* const* d_in, const int* in_sizes, int n_in,
                              void* d_out, int out_size, void* d_ws, size_t ws_size,
                              hipStream_t stream) {
    const float*     coords  = (const float*)d_in[0];
    const longlong2* pairs   = (const longlong2*)d_in[1];   // int64 [N_PAIRS, 2]
    const float*     box     = (const float*)d_in[2];
    const float*     sigma   = (const float*)d_in[3];
    const float*     epsilon = (const float*)d_in[4];
    const float*     charges = (const float*)d_in[5];
    const float*     coul    = (const float*)d_in[6];
    const float*     cutoff  = (const float*)d_in[7];

    const long long npairs  = (long long)in_sizes[1] / 2;
    const int       n_atoms = in_sizes[3];

    // Workspace layout: [0, BLOCKS*8) double partials; atoms table at 32 KiB.
    double* partials = (double*)d_ws;
    const size_t atoms_off = 32768;
    AtomPacked* atoms = (AtomPacked*)((char*)d_ws + atoms_off);
    const size_t need = atoms_off + (size_t)n_atoms * sizeof(AtomPacked);
    const bool use_packed = (ws_size >= need);

    if (use_packed) {
        nb_prep_kernel<<<(n_atoms + THREADS - 1) / THREADS, THREADS, 0, stream>>>(
            coords, sigma, epsilon, charges, atoms, n_atoms);
        nb_pair_kernel<true><<<BLOCKS, THREADS, 0, stream>>>(
            pairs, atoms, coords, sigma, epsilon, charges,
            box, coul, cutoff, npairs, partials);
    } else {
        nb_pair_kernel<false><<<BLOCKS, THREADS, 0, stream>>>(
            pairs, atoms, coords, sigma, epsilon, charges,
            box, coul, cutoff, npairs, partials);
    }

    nb_reduce_kernel<<<1, THREADS, 0, stream>>>(partials, BLOCKS, (float*)d_out);
}